// _ProposaLayer_49821620633809
// MI455X (gfx1250) — compile-verified
//
#include <hip/hip_runtime.h>
#include <hip/hip_bf16.h>

#define B_    16
#define A_    9
#define H_    192
#define W_    192
#define K_    (H_*W_)        /* 36864 */
#define N_    (A_*K_)        /* 331776 per batch */
#define PRE_  6000
#define POST_ 300
#define NMS_TH 0.7f
#define NEGV  (-1e9f)
#define NBIN  65536
#define NEG_BITS 0xCE6E6B28u /* bit pattern of -1e9f */

#define TRY_ASYNC 1   /* per-lane global_load_async_to_lds_* (ASYNCcnt path) */
#define TRY_TDM   1   /* tensor_load_to_lds for the contiguous box tile (TENSORcnt path) */

// py-faster-rcnn anchors, base 16, ratios {0.5,1,2}, scales {8,16,32}
__constant__ float ANC[9][4] = {
  { -84.f,  -40.f,  99.f,  55.f}, {-176.f,  -88.f, 191.f, 103.f}, {-360.f, -184.f, 375.f, 199.f},
  { -56.f,  -56.f,  71.f,  71.f}, {-120.f, -120.f, 135.f, 135.f}, {-248.f, -248.f, 263.f, 263.f},
  { -36.f,  -80.f,  51.f,  95.f}, { -80.f, -168.f,  95.f, 183.f}, {-168.f, -344.f, 183.f, 359.f}
};

__device__ __forceinline__ unsigned key_of(float s) {
  unsigned u = __float_as_uint(s);
  return (u & 0x80000000u) ? ~u : (u | 0x80000000u);  // monotonic float->uint
}

__global__ void k_fill(unsigned* p, unsigned v, unsigned n) {
  unsigned i = blockIdx.x * blockDim.x + threadIdx.x;
  if (i < n) p[i] = v;
}

// Pass 1 histogram: top 16 bits of score key, per batch.
__global__ void k_hist1(const float* __restrict__ sc, unsigned* __restrict__ hist1) {
  int idx = blockIdx.x * blockDim.x + threadIdx.x;
  if (idx >= B_ * N_) return;
  int b = idx / N_; int r = idx - b * N_;
  int a = r / K_;   int k = r - a * K_;
  float s = sc[(((size_t)b * (2*A_) + A_ + a) * K_) + k];
  atomicAdd(&hist1[(size_t)b * NBIN + (key_of(s) >> 16)], 1u);
}

// Pass 2 histogram: low 16 bits among elements whose hi16 == threshold bin.
__global__ void k_hist2(const float* __restrict__ sc, const unsigned* __restrict__ ctrl,
                        unsigned* __restrict__ hist2) {
  int idx = blockIdx.x * blockDim.x + threadIdx.x;
  if (idx >= B_ * N_) return;
  int b = idx / N_; int r = idx - b * N_;
  int a = r / K_;   int k = r - a * K_;
  float s = sc[(((size_t)b * (2*A_) + A_ + a) * K_) + k];
  unsigned key = key_of(s);
  if ((key >> 16) == ctrl[b*8 + 0])
    atomicAdd(&hist2[(size_t)b * NBIN + (key & 0xFFFFu)], 1u);
}

// Find hi16 threshold bin from top (256 threads/batch, chunked scan).
__global__ void k_sel1(const unsigned* __restrict__ hist1, unsigned* __restrict__ ctrl) {
  int b = blockIdx.x;
  const unsigned* h = hist1 + (size_t)b * NBIN;
  __shared__ unsigned cs[256];
  int c = threadIdx.x;
  unsigned s = 0;
  for (int i = 0; i < 256; ++i) s += h[c*256 + i];
  cs[c] = s;
  __syncthreads();
  if (c == 0) {
    unsigned want = PRE_, acc = 0; int ch = 255;
    for (; ch > 0; --ch) { if (acc + cs[ch] >= want) break; acc += cs[ch]; }
    int t1 = ch * 256;
    for (int i = 255; i >= 0; --i) {
      unsigned v = h[ch*256 + i];
      if (acc + v >= want) { t1 = ch*256 + i; break; }
      acc += v;
    }
    ctrl[b*8 + 0] = (unsigned)t1;   // hi16 threshold bin
    ctrl[b*8 + 1] = want - acc;     // remaining wanted inside bin t1
    ctrl[b*8 + 2] = acc;            // count strictly above t1
  }
}

// Find lo16 threshold, finalize exact 32-bit key threshold + counters.
__global__ void k_sel2(const unsigned* __restrict__ hist2, unsigned* __restrict__ ctrl) {
  int b = blockIdx.x;
  const unsigned* h = hist2 + (size_t)b * NBIN;
  __shared__ unsigned cs[256];
  int c = threadIdx.x;
  unsigned s = 0;
  for (int i = 0; i < 256; ++i) s += h[c*256 + i];
  cs[c] = s;
  __syncthreads();
  if (c == 0) {
    unsigned want = ctrl[b*8 + 1], acc = 0; int ch = 255;
    for (; ch > 0; --ch) { if (acc + cs[ch] >= want) break; acc += cs[ch]; }
    int t2 = ch * 256;
    for (int i = 255; i >= 0; --i) {
      unsigned v = h[ch*256 + i];
      if (acc + v >= want) { t2 = ch*256 + i; break; }
      acc += v;
    }
    unsigned cnt_gt = ctrl[b*8 + 2] + acc;           // exact count of keys > T (< PRE_)
    ctrl[b*8 + 3] = (ctrl[b*8 + 0] << 16) | (unsigned)t2;  // T
    ctrl[b*8 + 4] = cnt_gt;
    ctrl[b*8 + 5] = PRE_ - cnt_gt;                   // slots for keys == T
    ctrl[b*8 + 6] = 0;                               // gt counter
    ctrl[b*8 + 7] = 0;                               // eq counter
  }
}

// Compact top-PRE_ candidates; decode + clip boxes only for selected elements.
__global__ void k_compact(const float* __restrict__ sc_in, const float* __restrict__ dl_in,
                          const float* __restrict__ iminfo, unsigned* __restrict__ ctrl,
                          float* __restrict__ csc, float4* __restrict__ cbox) {
  int idx = blockIdx.x * blockDim.x + threadIdx.x;
  if (idx >= B_ * N_) return;
  int b = idx / N_; int r = idx - b * N_;
  int a = r / K_;   int k = r - a * K_;
  float s = sc_in[(((size_t)b * (2*A_) + A_ + a) * K_) + k];
  unsigned key = key_of(s);
  unsigned T = ctrl[b*8 + 3];
  int slot = -1;
  if (key > T) {
    slot = (int)atomicAdd(&ctrl[b*8 + 6], 1u);
  } else if (key == T) {
    unsigned p = atomicAdd(&ctrl[b*8 + 7], 1u);
    if (p < ctrl[b*8 + 5]) slot = (int)(ctrl[b*8 + 4] + p);
  }
  if (slot < 0) return;

  int w = k % W_, h = k / W_;
  float sx = w * 16.f, sy = h * 16.f;
  float x1 = ANC[a][0] + sx, y1 = ANC[a][1] + sy;
  float x2 = ANC[a][2] + sx, y2 = ANC[a][3] + sy;
  float aw = x2 - x1 + 1.f, ah = y2 - y1 + 1.f;
  float cx = x1 + 0.5f * aw, cy = y1 + 0.5f * ah;
  size_t dbase = (((size_t)b * (4*A_) + a*4) * K_) + k;
  float dx = dl_in[dbase];
  float dy = dl_in[dbase + (size_t)K_];
  float dw = dl_in[dbase + 2*(size_t)K_];
  float dh = dl_in[dbase + 3*(size_t)K_];
  float pcx = dx * aw + cx, pcy = dy * ah + cy;
  float pw = __expf(dw) * aw, ph = __expf(dh) * ah;
  float wmax = iminfo[b*3 + 1] - 1.f, hmax = iminfo[b*3 + 0] - 1.f;
  float bx1 = fminf(fmaxf(pcx - 0.5f*pw, 0.f), wmax);
  float by1 = fminf(fmaxf(pcy - 0.5f*ph, 0.f), hmax);
  float bx2 = fminf(fmaxf(pcx + 0.5f*pw, 0.f), wmax);
  float by2 = fminf(fmaxf(pcy + 0.5f*ph, 0.f), hmax);
  csc[(size_t)b * PRE_ + slot]  = s;
  cbox[(size_t)b * PRE_ + slot] = make_float4(bx1, by1, bx2, by2);
}

typedef const __attribute__((address_space(3))) char* lds_cptr_t;
__device__ __forceinline__ unsigned lds_off(const void* p) {
  // generic -> LDS addrspace -> 32-bit byte offset within workgroup LDS
  return (unsigned)(unsigned long long)(lds_cptr_t)p;
}

#if TRY_TDM
typedef unsigned uint32x4 __attribute__((ext_vector_type(4)));
typedef unsigned uint32x8 __attribute__((ext_vector_type(8)));
#endif

// Greedy NMS: one workgroup per batch; whole candidate pool LDS-resident (120 KB).
__global__ __launch_bounds__(1024)
void k_nms(const float* __restrict__ csc, const float4* __restrict__ cbox,
           float* __restrict__ out) {
  __shared__ float4 sbox[PRE_];   // 96 KB
  __shared__ float  ssc[PRE_];    // 24 KB
  __shared__ float  rsc[32];
  __shared__ int    ridx[32];
  __shared__ int    s_wi;
  __shared__ float  s_ws;

  int b = blockIdx.x, tid = threadIdx.x;
  const float4* gb = cbox + (size_t)b * PRE_;
  const float*  gs = csc  + (size_t)b * PRE_;

#if TRY_TDM
  // Stage the contiguous 96 KB box tile with one Tensor Data Mover op (wave 0 only).
  // D# per CDNA5 ISA sect. 8: 1-D tile, data_size=4B,
  // tensor_dim0 = tile_dim0 = PRE_*4 = 24000 elements, stride 24000.
  if (tid < 32) {
    unsigned long long ga = (unsigned long long)(const void*)gb;
    uint32x4 g0;
    g0[0] = 1u;                                              // count=1, user mode
    g0[1] = lds_off(&sbox[0]);                               // lds_addr (bytes)
    g0[2] = (unsigned)ga;                                    // global_addr[31:0]
    g0[3] = ((unsigned)(ga >> 32) & 0x01FFFFFFu) | (2u<<30); // addr[56:32] | type=2
    uint32x8 g1;
    g1[0] = 0x00020000u;  // workgroup_mask=0, data_size=2 (4B)
    g1[1] = 0x5DC00000u;  // tensor_dim0[15:0]=24000 in d1[31:16]
    g1[2] = 0x00010000u;  // tensor_dim0[31:16]=0, tensor_dim1=1
    g1[3] = 0x5DC00000u;  // tile_dim0=24000 in d3[31:16]
    g1[4] = 0x00000001u;  // tile_dim1=1, tile_dim2=0
    g1[5] = 24000u;       // tensor_dim0_stride[31:0]
    g1[6] = 0u;           // stride hi / tensor_dim1_stride lo
    g1[7] = 0u;
    uint32x4 gz = {0u, 0u, 0u, 0u};                          // groups 2/3 unused
    asm volatile("tensor_load_to_lds %0, %1, %2, %3"
                 :: "s"(g0), "s"(g1), "s"(gz), "s"(gz) : "memory");
    __builtin_amdgcn_s_wait_tensorcnt(0);
  }
#endif

  // Stage scores (and boxes, if no TDM) via per-lane async global->LDS loads.
  for (int i = tid; i < PRE_; i += 1024) {
#if TRY_ASYNC
#if !TRY_TDM
    unsigned d0 = lds_off(&sbox[i]);
    const void* a0 = (const void*)(gb + i);
    asm volatile("global_load_async_to_lds_b128 %0, %1, off"
                 :: "v"(d0), "v"(a0) : "memory");
#endif
    unsigned d1 = lds_off(&ssc[i]);
    const void* a1 = (const void*)(gs + i);
    asm volatile("global_load_async_to_lds_b32 %0, %1, off"
                 :: "v"(d1), "v"(a1) : "memory");
#else
#if !TRY_TDM
    sbox[i] = gb[i];
#endif
    ssc[i]  = gs[i];
#endif
  }
#if TRY_ASYNC
  asm volatile("s_wait_asynccnt 0" ::: "memory");
#endif
  __syncthreads();

  int lane = tid & 31, wv = tid >> 5;
  for (int t = 0; t < POST_; ++t) {
    // block-wide argmax (wave32 shuffle + 32-wave LDS reduce)
    float bs = -3e38f; int bi = 0;
    for (int i = tid; i < PRE_; i += 1024) {
      float s = ssc[i];
      if (s > bs) { bs = s; bi = i; }
    }
    for (int off = 16; off > 0; off >>= 1) {
      float os = __shfl_down(bs, off);
      int   oi = __shfl_down(bi, off);
      if (os > bs) { bs = os; bi = oi; }
    }
    if (lane == 0) { rsc[wv] = bs; ridx[wv] = bi; }
    __syncthreads();
    if (tid < 32) {
      float s2 = rsc[tid]; int i2 = ridx[tid];
      for (int off = 16; off > 0; off >>= 1) {
        float os = __shfl_down(s2, off);
        int   oi = __shfl_down(i2, off);
        if (os > s2) { s2 = os; i2 = oi; }
      }
      if (tid == 0) { s_wi = i2; s_ws = s2; }
    }
    __syncthreads();

    int wi = s_wi; float ws = s_ws;
    float4 wb = sbox[wi];
    if (tid == 0) {
      bool valid = ws > (NEGV * 0.5f);
      float* o = out + ((size_t)b * POST_ + t) * 5;
      o[0] = (float)b;                    // batch col always written (reference semantics)
      o[1] = valid ? wb.x : 0.f;
      o[2] = valid ? wb.y : 0.f;
      o[3] = valid ? wb.z : 0.f;
      o[4] = valid ? wb.w : 0.f;
    }
    // suppress everything with IoU > threshold against winner (includes winner itself)
    float wa = (wb.z - wb.x + 1.f) * (wb.w - wb.y + 1.f);
    for (int i = tid; i < PRE_; i += 1024) {
      float4 bx = sbox[i];
      float xx1 = fmaxf(wb.x, bx.x), yy1 = fmaxf(wb.y, bx.y);
      float xx2 = fminf(wb.z, bx.z), yy2 = fminf(wb.w, bx.w);
      float inter = fmaxf(xx2 - xx1 + 1.f, 0.f) * fmaxf(yy2 - yy1 + 1.f, 0.f);
      float ar = (bx.z - bx.x + 1.f) * (bx.w - bx.y + 1.f);
      float iou = inter / (wa + ar - inter);
      if (iou > NMS_TH) ssc[i] = NEGV;
    }
    __syncthreads();
  }
}

extern "C" void kernel_launch(void* const* d_in, const int* in_sizes, int n_in,
                              void* d_out, int out_size, void* d_ws, size_t ws_size,
                              hipStream_t stream) {
  const float* scores = (const float*)d_in[0];   // [16, 18, 192, 192]
  const float* deltas = (const float*)d_in[1];   // [16, 36, 192, 192]
  const float* iminfo = (const float*)d_in[2];   // [16, 3]
  float* out = (float*)d_out;                    // [16, 300, 5]

  // workspace layout
  unsigned* hist1 = (unsigned*)d_ws;                       // B * 65536
  unsigned* hist2 = hist1 + (size_t)B_ * NBIN;             // B * 65536
  unsigned* ctrl  = hist2 + (size_t)B_ * NBIN;             // B * 8
  float*    csc   = (float*)(ctrl + B_ * 8);               // B * PRE_
  float4*   cbox  = (float4*)(csc + (size_t)B_ * PRE_);    // B * PRE_ (16B aligned)

  const unsigned n_hist = 2u * B_ * NBIN;                  // 2,097,152
  const unsigned n_csc  = B_ * PRE_;                       // 96,000
  const unsigned n_cbox = B_ * PRE_ * 4;                   // 384,000
  k_fill<<<(n_hist + 255) / 256, 256, 0, stream>>>(hist1, 0u, n_hist);
  k_fill<<<(n_csc  + 255) / 256, 256, 0, stream>>>((unsigned*)csc,  NEG_BITS, n_csc);
  k_fill<<<(n_cbox + 255) / 256, 256, 0, stream>>>((unsigned*)cbox, 0u,       n_cbox);

  const int total  = B_ * N_;                              // 5,308,416
  const int blocks = (total + 255) / 256;
  k_hist1  <<<blocks, 256, 0, stream>>>(scores, hist1);
  k_sel1   <<<B_, 256, 0, stream>>>(hist1, ctrl);
  k_hist2  <<<blocks, 256, 0, stream>>>(scores, ctrl, hist2);
  k_sel2   <<<B_, 256, 0, stream>>>(hist2, ctrl);
  k_compact<<<blocks, 256, 0, stream>>>(scores, deltas, iminfo, ctrl, csc, cbox);
  k_nms    <<<B_, 1024, 0, stream>>>(csc, cbox, out);
}